// BVH_58239756534290
// MI455X (gfx1250) — compile-verified
//
#include <hip/hip_runtime.h>

// Closest-point-on-triangle brute force for MI455X (gfx1250).
// B=2, F=16384, Q=16384. Compute-bound on VALU (~30G ops); inputs (~1.6MB)
// are L2-resident, so HBM BW is irrelevant. We offload the per-pair dot
// products (d1, d2, n.ap, |ap|^2) onto the matrix pipe via
// V_WMMA_F32_16X16X4_F32 (K=4 affine dot with [p,1]), and keep the branchy
// Voronoi-region distance logic (division-free form) on VALU.

typedef __attribute__((ext_vector_type(2))) float v2f;
typedef __attribute__((ext_vector_type(8))) float v8f;

#define NB 2
#define NF 16384
#define NQ 16384
#define BQ (NB * NQ)

__device__ __forceinline__ float safercp(float x) {
    return (x != 0.0f) ? __builtin_amdgcn_rcpf(x) : 1.0f;
}
__device__ __forceinline__ float safediv(float n, float d) {
    return n / ((d != 0.0f) ? d : 1.0f);
}

__global__ __launch_bounds__(256) void
cpq_kernel(const float* __restrict__ tris,   // [NB][NF][3][3]
           const float* __restrict__ pts,    // [NB][NQ][3]
           float* __restrict__ out) {
    const int b    = blockIdx.y;
    const int wave = threadIdx.x >> 5;
    const int lane = threadIdx.x & 31;
    const int half = lane >> 4;    // 0: rows 0-7 / K 0-1, 1: rows 8-15 / K 2-3
    const int col  = lane & 15;    // triangle column (N) this lane owns
    const int pbase = blockIdx.x * 128 + wave * 16;  // first point of this wave

    // ---- load this lane's point (point pbase+col), build A fragment ----
    const float* pp = pts + ((size_t)b * NQ + pbase + col) * 3;
    const float px = pp[0], py = pp[1], pz = pp[2];
    const float mypp = px * px + py * py + pz * pz;

    // A is 16x4 (M x K), row m = [px,py,pz,1].
    // Lane layout: lanes 0-15 hold (K=0,K=1) of row M=lane; lanes 16-31 hold (K=2,K=3).
    v2f a_frag;
    a_frag.x = half ? pz : px;
    a_frag.y = half ? 1.0f : py;

    // Seed accumulator for |ap|^2 matrix with |p|^2 per row.
    // C/D layout: VGPR r holds row M=r (lanes 0-15) / M=r+8 (lanes 16-31).
    v8f pp_c;
#pragma unroll
    for (int r = 0; r < 8; ++r)
        pp_c[r] = __shfl(mypp, r + 8 * half, 32);

    float best[8];
    int bidx[8];
#pragma unroll
    for (int r = 0; r < 8; ++r) { best[r] = 3.402823466e+38f; bidx[r] = 0; }

    const float* tbase = tris + (size_t)b * NF * 9;

    for (int t = 0; t < NF; t += 16) {
        // ---- per-triangle constants (lane's triangle = t+col; both halves redundant) ----
        const float* tp = tbase + (size_t)(t + col) * 9;
        __builtin_prefetch(tp + 16 * 9, 0, 1);  // global_prefetch_b8, next tile
        const float ax = tp[0], ay = tp[1], az = tp[2];
        const float vbx = tp[3], vby = tp[4], vbz = tp[5];
        const float vcx = tp[6], vcy = tp[7], vcz = tp[8];
        const float abx = vbx - ax, aby = vby - ay, abz = vbz - az;
        const float acx = vcx - ax, acy = vcy - ay, acz = vcz - az;
        const float nx = aby * acz - abz * acy;
        const float ny = abz * acx - abx * acz;
        const float nz = abx * acy - aby * acx;
        const float ab2  = abx * abx + aby * aby + abz * abz;
        const float ac2  = acx * acx + acy * acy + acz * acz;
        const float abac = abx * acx + aby * acy + abz * acz;
        const float bc2  = ab2 - 2.0f * abac + ac2;
        const float n2   = nx * nx + ny * ny + nz * nz;
        const float aba  = abx * ax + aby * ay + abz * az;
        const float aca  = acx * ax + acy * ay + acz * az;
        const float na   = nx * ax + ny * ay + nz * az;
        const float a2   = ax * ax + ay * ay + az * az;
        const float iab2 = safercp(ab2), iac2 = safercp(ac2);
        const float ibc2 = safercp(bc2), in2  = safercp(n2);

        // B is 4x16 (K x N): lanes 0-15 hold (K=0,K=1) of col N=lane; lanes 16-31 (K=2,K=3).
        v2f b1, b2, b3, b4;
        b1.x = half ? abz : abx;            b1.y = half ? -aba : aby;   // d1 = ab.p - ab.a
        b2.x = half ? acz : acx;            b2.y = half ? -aca : acy;   // d2 = ac.p - ac.a
        b3.x = half ? nz : nx;              b3.y = half ? -na : ny;     // dn = n.p  - n.a
        b4.x = half ? (-2.0f * az) : (-2.0f * ax);
        b4.y = half ? a2 : (-2.0f * ay);                                // |a|^2 - 2 a.p (+|p|^2 via C)

        v8f zc = {};
        v8f C1 = __builtin_amdgcn_wmma_f32_16x16x4_f32(false, a_frag, false, b1, (short)0, zc,   false, false);
        v8f C2 = __builtin_amdgcn_wmma_f32_16x16x4_f32(false, a_frag, false, b2, (short)0, zc,   false, false);
        v8f C3 = __builtin_amdgcn_wmma_f32_16x16x4_f32(false, a_frag, false, b3, (short)0, zc,   false, false);
        v8f C4 = __builtin_amdgcn_wmma_f32_16x16x4_f32(false, a_frag, false, b4, (short)0, pp_c, false, false);

#pragma unroll
        for (int r = 0; r < 8; ++r) {
            const float d1 = C1[r], d2 = C2[r], dn = C3[r], psq = C4[r];
            const float d3 = d1 - ab2;
            const float d4 = d2 - abac;
            const float d5 = d1 - abac;
            const float d6 = d2 - ac2;
            const float vc = d1 * d4 - d3 * d2;
            const float vb = d5 * d2 - d1 * d6;
            const float va = d3 * d6 - d5 * d4;
            const float dB = psq - 2.0f * d1 + ab2;   // |bp|^2
            const float dC = psq - 2.0f * d2 + ac2;   // |cp|^2
            const float e  = d4 - d3;                 // bc.bp
            // division-free region distances; apply in reverse priority order
            float dist = dn * dn * in2;               // interior
            { const bool c = (va <= 0.f) & (e >= 0.f) & ((d5 - d6) >= 0.f);
              const float dd = dB - e * e * ibc2;          dist = c ? dd : dist; }   // edge BC
            { const bool c = (vb <= 0.f) & (d2 >= 0.f) & (d6 <= 0.f);
              const float dd = psq - d2 * d2 * iac2;       dist = c ? dd : dist; }   // edge AC
            { const bool c = (d6 >= 0.f) & (d5 <= d6);     dist = c ? dC : dist; }   // vertex C
            { const bool c = (vc <= 0.f) & (d1 >= 0.f) & (d3 <= 0.f);
              const float dd = psq - d1 * d1 * iab2;       dist = c ? dd : dist; }   // edge AB
            { const bool c = (d3 >= 0.f) & (d4 <= d3);     dist = c ? dB : dist; }   // vertex B
            { const bool c = (d1 <= 0.f) & (d2 <= 0.f);    dist = c ? psq : dist; }  // vertex A
            if (dist < best[r]) { best[r] = dist; bidx[r] = t + col; }
        }
    }

    // ---- cross-lane argmin over the 16 triangle columns (stay inside each half) ----
#pragma unroll
    for (int r = 0; r < 8; ++r) {
#pragma unroll
        for (int s = 1; s < 16; s <<= 1) {
            const float od = __shfl_xor(best[r], s, 32);
            const int   oi = __shfl_xor(bidx[r], s, 32);
            if (od < best[r] || (od == best[r] && oi < bidx[r])) {
                best[r] = od; bidx[r] = oi;
            }
        }
    }

    __shared__ int sIdx[8][16];
    if (col == 0) {
#pragma unroll
        for (int r = 0; r < 8; ++r) sIdx[wave][r + 8 * half] = bidx[r];
    }
    __syncthreads();

    // ---- refinement: exact reference math on the single winning triangle ----
    if (lane < 16) {
        const int q  = pbase + lane;
        const int fi = sIdx[wave][lane];
        const float* qp = pts + ((size_t)b * NQ + q) * 3;
        const float qx = qp[0], qy = qp[1], qz = qp[2];
        const float* tp = tbase + (size_t)fi * 9;
        const float ax = tp[0], ay = tp[1], az = tp[2];
        const float bx = tp[3], by = tp[4], bz = tp[5];
        const float cx = tp[6], cy = tp[7], cz = tp[8];
        const float abx = bx - ax, aby = by - ay, abz = bz - az;
        const float acx = cx - ax, acy = cy - ay, acz = cz - az;
        const float apx = qx - ax, apy = qy - ay, apz = qz - az;
        const float bpx = qx - bx, bpy = qy - by, bpz = qz - bz;
        const float cpx = qx - cx, cpy = qy - cy, cpz = qz - cz;
        const float d1 = abx * apx + aby * apy + abz * apz;
        const float d2 = acx * apx + acy * apy + acz * apz;
        const float d3 = abx * bpx + aby * bpy + abz * bpz;
        const float d4 = acx * bpx + acy * bpy + acz * bpz;
        const float d5 = abx * cpx + aby * cpy + abz * cpz;
        const float d6 = acx * cpx + acy * cpy + acz * cpz;
        const float vc = d1 * d4 - d3 * d2;
        const float vb = d5 * d2 - d1 * d6;
        const float va = d3 * d6 - d5 * d4;
        const float v_ab = safediv(d1, d1 - d3);
        const float w_ac = safediv(d2, d2 - d6);
        const float w_bc = safediv(d4 - d3, (d4 - d3) + (d5 - d6));
        float denom = va + vb + vc; denom = (denom != 0.0f) ? denom : 1.0f;
        const float v_in = vb / denom, w_in = vc / denom;
        float u = 1.0f - v_in - w_in, v = v_in, w = w_in;  // interior default
        if ((va <= 0.f) && ((d4 - d3) >= 0.f) && ((d5 - d6) >= 0.f)) { u = 0.f; v = 1.f - w_bc; w = w_bc; }
        if ((vb <= 0.f) && (d2 >= 0.f) && (d6 <= 0.f))               { u = 1.f - w_ac; v = 0.f; w = w_ac; }
        if ((d6 >= 0.f) && (d5 <= d6))                               { u = 0.f; v = 0.f; w = 1.f; }
        if ((vc <= 0.f) && (d1 >= 0.f) && (d3 <= 0.f))               { u = 1.f - v_ab; v = v_ab; w = 0.f; }
        if ((d3 >= 0.f) && (d4 <= d3))                               { u = 0.f; v = 1.f; w = 0.f; }
        if ((d1 <= 0.f) && (d2 <= 0.f))                              { u = 1.f; v = 0.f; w = 0.f; }
        const float gx = u * ax + v * bx + w * cx;
        const float gy = u * ay + v * by + w * cy;
        const float gz = u * az + v * bz + w * cz;
        const float dx = qx - gx, dy = qy - gy, dz = qz - gz;
        const float dist = dx * dx + dy * dy + dz * dz;
        const size_t g = (size_t)b * NQ + q;
        out[g] = dist;
        out[(size_t)BQ + 3 * g + 0] = gx;
        out[(size_t)BQ + 3 * g + 1] = gy;
        out[(size_t)BQ + 3 * g + 2] = gz;
        out[(size_t)4 * BQ + g] = (float)fi;
        out[(size_t)5 * BQ + 3 * g + 0] = u;
        out[(size_t)5 * BQ + 3 * g + 1] = v;
        out[(size_t)5 * BQ + 3 * g + 2] = w;
    }
}

extern "C" void kernel_launch(void* const* d_in, const int* in_sizes, int n_in,
                              void* d_out, int out_size, void* d_ws, size_t ws_size,
                              hipStream_t stream) {
    (void)in_sizes; (void)n_in; (void)out_size; (void)d_ws; (void)ws_size;
    const float* tris = (const float*)d_in[0];
    const float* pts  = (const float*)d_in[1];
    float* out        = (float*)d_out;
    dim3 grid(NQ / 128, NB);   // 128 blocks/batch, 8 waves * 16 points per block
    dim3 block(256);
    cpq_kernel<<<grid, block, 0, stream>>>(tris, pts, out);
}